// GQAAttention_11991548690955
// MI455X (gfx1250) — compile-verified
//
#include <hip/hip_runtime.h>
#include <math.h>

// ---------------- problem constants ----------------
#define D_      2048
#define H_      16
#define KH_     4
#define HD_     128
#define KVD_    512          // KH_ * HD_
#define GROUPS_ 4            // H_ / KH_
#define S_      2048
#define SP_     2048
#define B_      2
#define BS_     (B_ * S_)    // 4096
#define SOFTCAP_     30.0f
#define INV_SOFTCAP_ (1.0f / 30.0f)
#define LN_SCALE_    0.4082482904638631f   // 1/sqrt(6)
#define EPS_         1e-12f

typedef float v2f __attribute__((ext_vector_type(2)));
typedef float v8f __attribute__((ext_vector_type(8)));

__device__ __forceinline__ v8f v8f_zero() {
  v8f z = {0.f, 0.f, 0.f, 0.f, 0.f, 0.f, 0.f, 0.f};
  return z;
}

// Native f32 WMMA: D(16x16) += A(16x4) * B(4x16), f32 throughout.
__device__ __forceinline__ v8f wmma_f32(v2f a, v2f b, v8f c) {
  return __builtin_amdgcn_wmma_f32_16x16x4_f32(false, a, false, b, (short)0, c,
                                               false, false);
}

// Layout assumptions (CDNA5 ISA 7.12.2, 32-bit operands, wave32):
//  A (16x4):  lanes 0-15 hold M=lane; vgpr0={K0 | K2}, vgpr1={K1 | K3} by lane half
//  B (4x16):  lanes 0-15 hold N=lane; vgpr0={K0 | K2}, vgpr1={K1 | K3} by lane half
//  C/D(16x16):lane<16: vgpr r = [M=r][N=lane]; lane>=16: vgpr r = [M=r+8][N=lane-16]

// ================= generic f32 WMMA GEMM: C[M,N] = A[M,K] @ B[K,N] =============
// One wave computes a 16x64 tile. M,K mult of 16/4; N mult of 64.
__global__ __launch_bounds__(32) void gemm16x64(const float* __restrict__ A,
                                                const float* __restrict__ B,
                                                float* __restrict__ C,
                                                int M, int N, int K) {
  const int lane = threadIdx.x;
  const int row0 = blockIdx.y * 16;
  const int col0 = blockIdx.x * 64;
  const int mrow = lane & 15;
  const int ncol = lane & 15;
  const int koff = (lane >> 4) * 2;   // 0 or 2

  v8f acc[4];
#pragma unroll
  for (int c = 0; c < 4; ++c) acc[c] = v8f_zero();

  const float* arow = A + (size_t)(row0 + mrow) * K;
  for (int k = 0; k < K; k += 4) {
    v2f a = *(const v2f*)(arow + k + koff);            // one b64 load
    const float* bp0 = B + (size_t)(k + koff) * N + col0 + ncol;
    const float* bp1 = bp0 + N;
#pragma unroll
    for (int c = 0; c < 4; ++c) {
      v2f b = {bp0[c * 16], bp1[c * 16]};
      acc[c] = wmma_f32(a, b, acc[c]);
    }
  }
#pragma unroll
  for (int c = 0; c < 4; ++c) {
#pragma unroll
    for (int r = 0; r < 8; ++r) {
      int rr = (lane < 16) ? r : (r + 8);
      C[(size_t)(row0 + rr) * N + col0 + c * 16 + ncol] = acc[c][r];
    }
  }
}

// ============ row-wise L2 norm over HD=128, optional per-head gain ============
// One wave per row; 4 waves per block.
__global__ __launch_bounds__(128) void l2norm_rows(float* __restrict__ buf,
                                                   const float* __restrict__ gain,
                                                   int nheads, float gscale,
                                                   int use_gain, int nrows) {
  int rid = blockIdx.x * 4 + (threadIdx.x >> 5);
  int lane = threadIdx.x & 31;
  if (rid >= nrows) return;
  float* row = buf + (size_t)rid * HD_;
  float4 v = *(const float4*)(row + lane * 4);
  float ss = v.x * v.x + v.y * v.y + v.z * v.z + v.w * v.w;
#pragma unroll
  for (int m = 16; m >= 1; m >>= 1) ss += __shfl_xor(ss, m, 32);
  float scale = 1.0f / fmaxf(sqrtf(ss), EPS_);
  if (use_gain) scale *= gain[rid % nheads] * gscale;
  v.x *= scale; v.y *= scale; v.z *= scale; v.w *= scale;
  *(float4*)(row + lane * 4) = v;
}

// ======================= flash attention, one wave per ========================
// (batch, head, 16-row query tile). K-dim of scores = HD = 128 (32 WMMAs);
// P·V uses 4 WMMAs per 16-col output chunk x 8 chunks. KV concat handled by
// per-tile pointer selection (tiles are 16-aligned, SP_ is a multiple of 16).
__global__ __launch_bounds__(32) void attn16(const float* __restrict__ qbuf,  // [B,S,H,HD]
                                             const float* __restrict__ kbuf,  // [B,S,KH,HD]
                                             const float* __restrict__ vbuf,  // [B,S,KH,HD]
                                             const float* __restrict__ prev_k,// [B,SP,KH,HD]
                                             const float* __restrict__ prev_v,
                                             float* __restrict__ obuf) {      // [B,S,H,HD]
  __shared__ float pl[16 * 16];
  const int lane = threadIdx.x;
  const int qt = blockIdx.x;
  const int h  = blockIdx.y;
  const int b  = blockIdx.z;
  const int kh = h / GROUPS_;           // repeat_interleave GQA mapping
  const int qbase = qt * 16;
  const int mrow = lane & 15;
  const int ncol = lane & 15;
  const int koff = (lane >> 4) * 2;

  // Q tile (16 x 128) in A-fragment registers.
  v2f qa[32];
  const float* qrow = qbuf + (size_t)(b * S_ + qbase + mrow) * D_ + h * HD_;
#pragma unroll
  for (int kc = 0; kc < 32; ++kc)
    qa[kc] = *(const v2f*)(qrow + kc * 4 + koff);

  const float* Kprev = prev_k + ((size_t)b * SP_ * KH_ + kh) * HD_;  // j-stride KVD_
  const float* Vprev = prev_v + ((size_t)b * SP_ * KH_ + kh) * HD_;
  const float* Knew  = kbuf + (size_t)b * S_ * KVD_ + kh * HD_;
  const float* Vnew  = vbuf + (size_t)b * S_ * KVD_ + kh * HD_;

  float mst[8], lst[8];
  v8f oacc[8];
#pragma unroll
  for (int r = 0; r < 8; ++r) { mst[r] = -3.0e38f; lst[r] = 0.0f; }
#pragma unroll
  for (int c = 0; c < 8; ++c) oacc[c] = v8f_zero();

  // Allowed keys: j <= i + SP_. Only the tile starting at qbase+SP_ is partial.
  const int ntiles = qt + SP_ / 16 + 1;
  for (int t = 0; t < ntiles; ++t) {
    const float *Kb, *Vb;
    if (t < SP_ / 16) {
      Kb = Kprev + (size_t)t * 16 * KVD_;
      Vb = Vprev + (size_t)t * 16 * KVD_;
    } else {
      Kb = Knew + (size_t)(t * 16 - SP_) * KVD_;
      Vb = Vnew + (size_t)(t * 16 - SP_) * KVD_;
    }

    // ---- S = Q K^T (16x16 tile, K-dim 128) ----
    v8f s = v8f_zero();
    const float* krow = Kb + (size_t)ncol * KVD_;   // key = t*16 + ncol
#pragma unroll
    for (int kc = 0; kc < 32; ++kc) {
      v2f bb = *(const v2f*)(krow + kc * 4 + koff);
      s = wmma_f32(qa[kc], bb, s);
    }

    // ---- soft-cap, mask, online softmax (rows map to lane halves) ----
    const bool last = (t == ntiles - 1);
    float pr[8];
#pragma unroll
    for (int r = 0; r < 8; ++r) {
      float sv = SOFTCAP_ * tanhf(s[r] * INV_SOFTCAP_);
      if (last) {
        int irow = qbase + ((lane < 16) ? r : r + 8);
        int j = t * 16 + ncol;
        if (j > irow + SP_) sv = -1.0e9f;
      }
      float rmax = sv;
      rmax = fmaxf(rmax, __shfl_xor(rmax, 1, 32));
      rmax = fmaxf(rmax, __shfl_xor(rmax, 2, 32));
      rmax = fmaxf(rmax, __shfl_xor(rmax, 4, 32));
      rmax = fmaxf(rmax, __shfl_xor(rmax, 8, 32));
      float mnew = fmaxf(mst[r], rmax);
      float corr = expf(mst[r] - mnew);
      float p = expf(sv - mnew);
      float rs = p;
      rs += __shfl_xor(rs, 1, 32);
      rs += __shfl_xor(rs, 2, 32);
      rs += __shfl_xor(rs, 4, 32);
      rs += __shfl_xor(rs, 8, 32);
      lst[r] = lst[r] * corr + rs;
      mst[r] = mnew;
#pragma unroll
      for (int c = 0; c < 8; ++c) oacc[c][r] *= corr;
      pr[r] = p;
    }

    // ---- D-layout -> A-layout for P via LDS (single wave; dscnt fence) ----
#pragma unroll
    for (int r = 0; r < 8; ++r) {
      int row = (lane < 16) ? r : r + 8;
      pl[row * 16 + ncol] = pr[r];
    }
    asm volatile("s_wait_dscnt 0" ::: "memory");
    v2f pa[4];
#pragma unroll
    for (int kc = 0; kc < 4; ++kc)
      pa[kc] = *(const v2f*)(&pl[mrow * 16 + kc * 4 + koff]);
    asm volatile("s_wait_dscnt 0" ::: "memory");  // reads done before next overwrite

    // ---- O += P V (16 x 128) ----
#pragma unroll
    for (int c = 0; c < 8; ++c) {
#pragma unroll
      for (int kc = 0; kc < 4; ++kc) {
        const float* vp = Vb + (size_t)(kc * 4 + koff) * KVD_ + c * 16 + ncol;
        v2f bb = {vp[0], vp[KVD_]};
        oacc[c] = wmma_f32(pa[kc], bb, oacc[c]);
      }
    }
  }

  // ---- epilogue: O / l ----
  float* orow = obuf + (size_t)(b * S_ + qbase) * D_ + h * HD_;
#pragma unroll
  for (int c = 0; c < 8; ++c) {
#pragma unroll
    for (int r = 0; r < 8; ++r) {
      int row = (lane < 16) ? r : r + 8;
      orow[(size_t)row * D_ + c * 16 + ncol] = oacc[c][r] / lst[r];
    }
  }
}

// ================================ launcher ====================================
extern "C" void kernel_launch(void* const* d_in, const int* in_sizes, int n_in,
                              void* d_out, int out_size, void* d_ws, size_t ws_size,
                              hipStream_t stream) {
  (void)in_sizes; (void)n_in; (void)out_size; (void)ws_size;
  const float* x      = (const float*)d_in[0];
  const float* prev_k = (const float*)d_in[1];
  const float* prev_v = (const float*)d_in[2];
  const float* Wq     = (const float*)d_in[3];
  const float* Wk     = (const float*)d_in[4];
  const float* Wv     = (const float*)d_in[5];
  const float* Wo     = (const float*)d_in[6];
  const float* qk_gain = (const float*)d_in[7];
  float* out = (float*)d_out;

  // Workspace: q (32MB) | k (8MB) | v (8MB) | attn_out (32MB) = 80 MB
  float* qbuf = (float*)d_ws;
  float* kbuf = qbuf + (size_t)BS_ * D_;
  float* vbuf = kbuf + (size_t)BS_ * KVD_;
  float* abuf = vbuf + (size_t)BS_ * KVD_;

  dim3 wave(32);
  // Projections
  gemm16x64<<<dim3(D_ / 64,   BS_ / 16), wave, 0, stream>>>(x, Wq, qbuf, BS_, D_,   D_);
  gemm16x64<<<dim3(KVD_ / 64, BS_ / 16), wave, 0, stream>>>(x, Wk, kbuf, BS_, KVD_, D_);
  gemm16x64<<<dim3(KVD_ / 64, BS_ / 16), wave, 0, stream>>>(x, Wv, vbuf, BS_, KVD_, D_);
  // QK norm (+gain*LN_SCALE on Q)
  l2norm_rows<<<(BS_ * H_) / 4,  128, 0, stream>>>(qbuf, qk_gain, H_,  LN_SCALE_, 1, BS_ * H_);
  l2norm_rows<<<(BS_ * KH_) / 4, 128, 0, stream>>>(kbuf, qk_gain, KH_, 1.0f,      0, BS_ * KH_);
  // Flash attention
  attn16<<<dim3(S_ / 16, H_, B_), wave, 0, stream>>>(qbuf, kbuf, vbuf, prev_k, prev_v, abuf);
  // Output projection
  gemm16x64<<<dim3(D_ / 64, BS_ / 16), wave, 0, stream>>>(abuf, Wo, out, BS_, D_, D_);
}